// Momentum_Transform_68040871903237
// MI455X (gfx1250) — compile-verified
//
#include <hip/hip_runtime.h>
#include <cstdint>
#include <cstddef>

#define B_SZ   8
#define N_PTS  4096
#define KNN    20
#define BN_EPS 1e-5f
#define NEG_INF (-3.402823466e38f)

typedef float v2f __attribute__((ext_vector_type(2)));
typedef float v8f __attribute__((ext_vector_type(8)));

#if defined(__HIP_DEVICE_COMPILE__) && __has_builtin(__builtin_amdgcn_wmma_f32_16x16x4_f32)
#define WMMA_F32_OK 1
#else
#define WMMA_F32_OK 0
#endif

// ---------------------------------------------------------------------------
// Workspace layout (offsets in floats)
// ---------------------------------------------------------------------------
static constexpr size_t OFF_A    = 0;        // 8*64*4096  (64-ch buffer #1)
static constexpr size_t OFF_B    = 2097152;  // 8*128*4096 (128-ch buffer)
static constexpr size_t OFF_C    = 6291456;  // 8*64*4096  (64-ch buffer #2)
static constexpr size_t OFF_PTS  = 8388608;  // 8*4096*3
static constexpr size_t OFF_XX   = 8486912;  // 8*4096
static constexpr size_t OFF_POOL = 8519680;  // 8*1024
static constexpr size_t OFF_F1Z  = 8527872;  // 8*512
static constexpr size_t OFF_F1   = 8531968;  // 8*512
static constexpr size_t OFF_F2Z  = 8536064;  // 8*256
static constexpr size_t OFF_F2   = 8538112;  // 8*256
static constexpr size_t OFF_MAT  = 8540160;  // 8*9 (pad)
static constexpr size_t OFF_STAT = 8540240;  // 2*1024 (sum, sumsq)
static constexpr size_t OFF_SC   = 8542288;  // 1024
static constexpr size_t OFF_SH   = 8543312;  // 1024
static constexpr size_t OFF_S1   = 8544336;  // 12 + 78 contiguous moments
static constexpr size_t OFF_IDX  = 8544432;  // 8*4096*20 ints

// ---------------------------------------------------------------------------
// WMMA 16x16 fp32 tile:  D(16x16) += Wtile(16xC) * X(CxN) over columns n0..n0+15
// Layout (ISA 7.12.2):
//   A 16x4 f32 : lane l (l<16): rows l, VGPR0=K0, VGPR1=K1; lanes 16-31: K2,K3
//   C/D 16x16  : VGPR v: lanes 0-15 row v (col=lane), lanes 16-31 row v+8
// Scalar fallback reproduces the exact same register layout so that all
// downstream stores / reductions are shared between paths.
// ---------------------------------------------------------------------------
__device__ __forceinline__ void gemm_tile16(const float* __restrict__ Wl,   // LDS 16 x C
                                            const float* __restrict__ Xb,   // C x N (one batch)
                                            int n0, int C, float acc[8]) {
  const int lane = threadIdx.x & 31;
  const int half = lane >> 4;
  const int l15  = lane & 15;
#if WMMA_F32_OK
  v8f c;
#pragma unroll
  for (int v = 0; v < 8; ++v) c[v] = acc[v];
  const int ko = half * 2;
  for (int k = 0; k < C; k += 4) {
    v2f a, b;
    a.x = Wl[l15 * C + k + ko];
    a.y = Wl[l15 * C + k + ko + 1];
    const float* xp = Xb + (size_t)(k + ko) * N_PTS + n0 + l15;
    b.x = xp[0];
    b.y = xp[N_PTS];
    c = __builtin_amdgcn_wmma_f32_16x16x4_f32(false, a, false, b, (short)0, c,
                                              false, false);
  }
#pragma unroll
  for (int v = 0; v < 8; ++v) acc[v] = c[v];
#else
#pragma unroll
  for (int v = 0; v < 8; ++v) {
    const int row = v + 8 * half;
    float s = acc[v];
    for (int k = 0; k < C; ++k)
      s += Wl[row * C + k] * Xb[(size_t)k * N_PTS + n0 + l15];
    acc[v] = s;
  }
#endif
}

// ---------------------------------------------------------------------------
// GEMM kernels (4 waves / block). grid handles O-tiles and batch.
// ---------------------------------------------------------------------------
__global__ __launch_bounds__(128) void mt_gemm_store(
    const float* __restrict__ W, const float* __restrict__ X,
    const float* __restrict__ bias, float* __restrict__ Y, int C, int O) {
  __shared__ float Wl[16 * 128];
  const int b = blockIdx.z, o0 = blockIdx.y * 16;
  const int wave = threadIdx.x >> 5;
  const int n0 = blockIdx.x * 64 + wave * 16;
  for (int i = threadIdx.x; i < 16 * C; i += 128)
    Wl[i] = W[(size_t)(o0 + i / C) * C + (i % C)];
  __syncthreads();
  float acc[8] = {0, 0, 0, 0, 0, 0, 0, 0};
  gemm_tile16(Wl, X + (size_t)b * C * N_PTS, n0, C, acc);
  const int lane = threadIdx.x & 31, half = lane >> 4, l15 = lane & 15;
#pragma unroll
  for (int v = 0; v < 8; ++v) {
    const int o = o0 + v + 8 * half;
    Y[((size_t)b * O + o) * N_PTS + n0 + l15] = acc[v] + bias[o];
  }
}

// Accumulate per-channel sum / sumsq (y = Wx + bias) without storing y.
__global__ __launch_bounds__(128) void mt_gemm_stats(
    const float* __restrict__ W, const float* __restrict__ X,
    const float* __restrict__ bias, float* __restrict__ stat, int C, int O) {
  __shared__ float Wl[16 * 128];
  const int b = blockIdx.y, o0 = blockIdx.x * 16;
  const int wave = threadIdx.x >> 5;
  for (int i = threadIdx.x; i < 16 * C; i += 128)
    Wl[i] = W[(size_t)(o0 + i / C) * C + (i % C)];
  __syncthreads();
  const int lane = threadIdx.x & 31, half = lane >> 4, l15 = lane & 15;
  float bv[8];
#pragma unroll
  for (int v = 0; v < 8; ++v) bv[v] = bias[o0 + v + 8 * half];
  float ls[8] = {0, 0, 0, 0, 0, 0, 0, 0}, lq[8] = {0, 0, 0, 0, 0, 0, 0, 0};
  for (int tile = 0; tile < N_PTS / 64; ++tile) {
    const int n0 = tile * 64 + wave * 16;
    float acc[8] = {0, 0, 0, 0, 0, 0, 0, 0};
    gemm_tile16(Wl, X + (size_t)b * C * N_PTS, n0, C, acc);
#pragma unroll
    for (int v = 0; v < 8; ++v) {
      const float t = acc[v] + bv[v];
      ls[v] += t;
      lq[v] += t * t;
    }
  }
#pragma unroll
  for (int v = 0; v < 8; ++v)
    for (int off = 1; off < 16; off <<= 1) {
      ls[v] += __shfl_xor(ls[v], off, 32);
      lq[v] += __shfl_xor(lq[v], off, 32);
    }
  if (l15 == 0) {
#pragma unroll
    for (int v = 0; v < 8; ++v) {
      const int o = o0 + v + 8 * half;
      atomicAdd(&stat[o], ls[v]);
      atomicAdd(&stat[O + o], lq[v]);
    }
  }
}

// y = Wx + bias, optional BN(scale/shift)+relu, then max over all N -> out[b*O+o]
__global__ __launch_bounds__(128) void mt_gemm_reduce_max(
    const float* __restrict__ W, const float* __restrict__ X,
    const float* __restrict__ bias, const float* __restrict__ sc,
    const float* __restrict__ sh, int useBn, int useRelu,
    float* __restrict__ out, int C, int O) {
  __shared__ float Wl[16 * 128];
  __shared__ float red[4][16];
  const int b = blockIdx.y, o0 = blockIdx.x * 16;
  const int wave = threadIdx.x >> 5;
  for (int i = threadIdx.x; i < 16 * C; i += 128)
    Wl[i] = W[(size_t)(o0 + i / C) * C + (i % C)];
  __syncthreads();
  const int lane = threadIdx.x & 31, half = lane >> 4, l15 = lane & 15;
  float bv[8], scv[8], shv[8];
#pragma unroll
  for (int v = 0; v < 8; ++v) {
    const int o = o0 + v + 8 * half;
    bv[v] = bias[o];
    scv[v] = useBn ? sc[o] : 1.f;
    shv[v] = useBn ? sh[o] : 0.f;
  }
  float rm[8];
#pragma unroll
  for (int v = 0; v < 8; ++v) rm[v] = NEG_INF;
  for (int tile = 0; tile < N_PTS / 64; ++tile) {
    const int n0 = tile * 64 + wave * 16;
    float acc[8] = {0, 0, 0, 0, 0, 0, 0, 0};
    gemm_tile16(Wl, X + (size_t)b * C * N_PTS, n0, C, acc);
#pragma unroll
    for (int v = 0; v < 8; ++v) {
      float t = acc[v] + bv[v];
      t = t * scv[v] + shv[v];
      if (useRelu) t = fmaxf(t, 0.f);
      rm[v] = fmaxf(rm[v], t);
    }
  }
#pragma unroll
  for (int v = 0; v < 8; ++v)
    for (int off = 1; off < 16; off <<= 1)
      rm[v] = fmaxf(rm[v], __shfl_xor(rm[v], off, 32));
  if (l15 == 0) {
#pragma unroll
    for (int v = 0; v < 8; ++v) red[wave][v + 8 * half] = rm[v];
  }
  __syncthreads();
  if (threadIdx.x < 16) {
    float m = red[0][threadIdx.x];
    for (int w = 1; w < 4; ++w) m = fmaxf(m, red[w][threadIdx.x]);
    out[(size_t)b * O + o0 + threadIdx.x] = m;
  }
}

// ---------------------------------------------------------------------------
// Small / elementwise kernels
// ---------------------------------------------------------------------------
__global__ void mt_conv_small(const float* __restrict__ X,
                              const float* __restrict__ W,
                              const float* __restrict__ bias,
                              float* __restrict__ Y, int C, int O) {
  const size_t total = (size_t)B_SZ * O * N_PTS;
  for (size_t i = blockIdx.x * blockDim.x + threadIdx.x; i < total;
       i += (size_t)gridDim.x * blockDim.x) {
    const int n = (int)(i % N_PTS);
    const int o = (int)((i / N_PTS) % O);
    const int b = (int)(i / ((size_t)O * N_PTS));
    float acc = bias[o];
    for (int c = 0; c < C; ++c)
      acc += W[o * C + c] * X[((size_t)b * C + c) * N_PTS + n];
    Y[i] = acc;
  }
}

// per-channel mean/var over (B,N) of stored Y -> scale/shift
__global__ __launch_bounds__(256) void mt_stats_finalize(
    const float* __restrict__ Y, const float* __restrict__ gamma,
    const float* __restrict__ beta, float* __restrict__ sc,
    float* __restrict__ sh, int O) {
  __shared__ float ss[256], sq[256];
  const int o = blockIdx.x;
  float s = 0.f, q = 0.f;
  for (int b = 0; b < B_SZ; ++b) {
    const float* row = Y + ((size_t)b * O + o) * N_PTS;
    for (int n = threadIdx.x; n < N_PTS; n += 256) {
      const float v = row[n];
      s += v;
      q += v * v;
    }
  }
  ss[threadIdx.x] = s;
  sq[threadIdx.x] = q;
  __syncthreads();
  for (int st = 128; st > 0; st >>= 1) {
    if (threadIdx.x < st) {
      ss[threadIdx.x] += ss[threadIdx.x + st];
      sq[threadIdx.x] += sq[threadIdx.x + st];
    }
    __syncthreads();
  }
  if (threadIdx.x == 0) {
    const float cnt = (float)(B_SZ * N_PTS);
    const float mean = ss[0] / cnt;
    const float var = sq[0] / cnt - mean * mean;
    const float r = gamma[o] * rsqrtf(var + BN_EPS);
    sc[o] = r;
    sh[o] = beta[o] - mean * r;
  }
}

__global__ void mt_bn_relu(float* __restrict__ Y, const float* __restrict__ sc,
                           const float* __restrict__ sh, int O) {
  const size_t total = (size_t)B_SZ * O * N_PTS;
  for (size_t i = blockIdx.x * blockDim.x + threadIdx.x; i < total;
       i += (size_t)gridDim.x * blockDim.x) {
    const int o = (int)((i / N_PTS) % O);
    Y[i] = fmaxf(Y[i] * sc[o] + sh[o], 0.f);
  }
}

__global__ void mt_finalize_sums(const float* __restrict__ stat,
                                 const float* __restrict__ gamma,
                                 const float* __restrict__ beta,
                                 float* __restrict__ sc, float* __restrict__ sh,
                                 int O, float cnt) {
  const int o = blockIdx.x * blockDim.x + threadIdx.x;
  if (o >= O) return;
  const float mean = stat[o] / cnt;
  const float var = stat[O + o] / cnt - mean * mean;
  const float r = gamma[o] * rsqrtf(var + BN_EPS);
  sc[o] = r;
  sh[o] = beta[o] - mean * r;
}

__global__ void mt_fc(const float* __restrict__ In, const float* __restrict__ W,
                      const float* __restrict__ bias, float* __restrict__ Out,
                      int Cin, int O, int addI) {
  const int total = B_SZ * O;
  for (int i = blockIdx.x * blockDim.x + threadIdx.x; i < total;
       i += gridDim.x * blockDim.x) {
    const int b = i / O, o = i % O;
    float acc = bias[o];
    for (int k = 0; k < Cin; ++k) acc += In[b * Cin + k] * W[o * Cin + k];
    if (addI && (o == 0 || o == 4 || o == 8)) acc += 1.f;
    Out[i] = acc;
  }
}

// BN over batch axis (8 samples) per channel, + relu
__global__ void mt_bn1d_relu(const float* __restrict__ Z,
                             const float* __restrict__ gamma,
                             const float* __restrict__ beta,
                             float* __restrict__ F, int O) {
  const int o = blockIdx.x * blockDim.x + threadIdx.x;
  if (o >= O) return;
  float s = 0.f, q = 0.f;
  for (int b = 0; b < B_SZ; ++b) {
    const float v = Z[b * O + o];
    s += v;
    q += v * v;
  }
  const float mean = s / B_SZ;
  const float var = q / B_SZ - mean * mean;
  const float r = gamma[o] * rsqrtf(var + BN_EPS);
  const float shf = beta[o] - mean * r;
  for (int b = 0; b < B_SZ; ++b)
    F[b * O + o] = fmaxf(Z[b * O + o] * r + shf, 0.f);
}

// xb = einsum('bcn,bcd->bdn'); write pts (B,N,3) + squared norms xx (B,N)
__global__ void mt_transform(const float* __restrict__ x,
                             const float* __restrict__ mat,
                             float* __restrict__ pts, float* __restrict__ xx) {
  const int total = B_SZ * N_PTS;
  for (int i = blockIdx.x * blockDim.x + threadIdx.x; i < total;
       i += gridDim.x * blockDim.x) {
    const int b = i / N_PTS, n = i % N_PTS;
    const float x0 = x[((size_t)b * 3 + 0) * N_PTS + n];
    const float x1 = x[((size_t)b * 3 + 1) * N_PTS + n];
    const float x2 = x[((size_t)b * 3 + 2) * N_PTS + n];
    const float* m = mat + b * 9;
    const float d0 = x0 * m[0] + x1 * m[3] + x2 * m[6];
    const float d1 = x0 * m[1] + x1 * m[4] + x2 * m[7];
    const float d2 = x0 * m[2] + x1 * m[5] + x2 * m[8];
    pts[(size_t)i * 3 + 0] = d0;
    pts[(size_t)i * 3 + 1] = d1;
    pts[(size_t)i * 3 + 2] = d2;
    xx[i] = d0 * d0 + d1 * d1 + d2 * d2;
  }
}

// top-K of neg_dist per row; per-thread sorted top-20, merged by 20 argmax rounds
__global__ __launch_bounds__(128) void mt_knn(const float* __restrict__ pts,
                                              const float* __restrict__ xx,
                                              int* __restrict__ idx) {
  const int row = blockIdx.x;           // b*N + i
  const int b = row / N_PTS;
  const int i = row % N_PTS;
  const int t = threadIdx.x;
  const float* pb = pts + (size_t)b * N_PTS * 3;
  const float pix = pb[(size_t)i * 3 + 0];
  const float piy = pb[(size_t)i * 3 + 1];
  const float piz = pb[(size_t)i * 3 + 2];
  const float xi = xx[row];
  float bv[KNN];
  int bi[KNN];
#pragma unroll
  for (int q = 0; q < KNN; ++q) { bv[q] = NEG_INF; bi[q] = -1; }
  for (int j = t; j < N_PTS; j += 128) {
    const float d = 2.f * (pix * pb[(size_t)j * 3] + piy * pb[(size_t)j * 3 + 1] +
                           piz * pb[(size_t)j * 3 + 2]) -
                    xi - xx[b * N_PTS + j];
    if (d > bv[KNN - 1]) {
      int q = KNN - 1;
      while (q > 0 && bv[q - 1] < d) {
        bv[q] = bv[q - 1];
        bi[q] = bi[q - 1];
        --q;
      }
      bv[q] = d;
      bi[q] = j;
    }
  }
  __shared__ float rv[128];
  __shared__ int rt[128];
  int ptr = 0;
  for (int round = 0; round < KNN; ++round) {
    rv[t] = (ptr < KNN) ? bv[ptr] : NEG_INF;
    rt[t] = t;
    __syncthreads();
    for (int s = 64; s > 0; s >>= 1) {
      if (t < s && rv[t + s] > rv[t]) { rv[t] = rv[t + s]; rt[t] = rt[t + s]; }
      __syncthreads();
    }
    const int w = rt[0];
    __syncthreads();
    if (t == w) {
      idx[(size_t)row * KNN + round] = bi[ptr];
      ++ptr;
    }
    __syncthreads();
  }
}

__device__ __forceinline__ void make_feat(const float* c3, const float* n3,
                                          float f[12]) {
  const float cx = c3[0], cy = c3[1], cz = c3[2];
  f[0] = cx; f[1] = cy; f[2] = cz;
  f[3] = cx * cx; f[4] = cy * cy; f[5] = cz * cz;
  f[6] = cx * cy; f[7] = cx * cz; f[8] = cy * cz;
  f[9] = n3[0] - cx; f[10] = n3[1] - cy; f[11] = n3[2] - cz;
}

// first & second moments of the 12-d feature over all (b,n,k); S[0..11]=Σf,
// S[12..89]=upper-tri Σ f_a f_b  (analytic BN stats for the 12->64 conv)
__global__ __launch_bounds__(256) void mt_feat_stats(
    const float* __restrict__ pts, const int* __restrict__ idx,
    float* __restrict__ S) {
  float s1[12];
  float s2[78];
#pragma unroll
  for (int a = 0; a < 12; ++a) s1[a] = 0.f;
#pragma unroll
  for (int a = 0; a < 78; ++a) s2[a] = 0.f;
  const size_t M = (size_t)B_SZ * N_PTS * KNN;
  for (size_t e = blockIdx.x * blockDim.x + threadIdx.x; e < M;
       e += (size_t)gridDim.x * blockDim.x) {
    const int k = (int)(e % KNN);
    (void)k;
    const size_t rowk = e / KNN;         // b*N + n
    const int b = (int)(rowk / N_PTS);
    const int j = idx[e];
    const float* c3 = pts + rowk * 3;
    const float* n3 = pts + ((size_t)b * N_PTS + j) * 3;
    float f[12];
    make_feat(c3, n3, f);
    int p = 0;
#pragma unroll
    for (int a = 0; a < 12; ++a) {
      s1[a] += f[a];
#pragma unroll
      for (int c = 0; c < 12; ++c)
        if (c >= a) s2[p + (c - a)] += f[a] * f[c];
      p += 12 - a;
    }
  }
  __shared__ float ls[90];
  if (threadIdx.x < 90) ls[threadIdx.x] = 0.f;
  __syncthreads();
#pragma unroll
  for (int a = 0; a < 12; ++a) atomicAdd(&ls[a], s1[a]);
#pragma unroll
  for (int a = 0; a < 78; ++a) atomicAdd(&ls[12 + a], s2[a]);
  __syncthreads();
  if (threadIdx.x < 90) atomicAdd(&S[threadIdx.x], ls[threadIdx.x]);
}

// derive per-channel BN scale/shift of y = cw1·f + cb1 from feature moments
__global__ void mt_graph_finalize(const float* __restrict__ w,
                                  const float* __restrict__ bias,
                                  const float* __restrict__ gamma,
                                  const float* __restrict__ beta,
                                  const float* __restrict__ S,
                                  float* __restrict__ sc,
                                  float* __restrict__ sh) {
  const int o = threadIdx.x;
  if (o >= 64) return;
  const float* wo = w + o * 12;
  const float be = bias[o];
  const float Mf = (float)((size_t)B_SZ * N_PTS * KNN);
  float s1w = 0.f;
  for (int a = 0; a < 12; ++a) s1w += wo[a] * S[a];
  float quad = 0.f;
  int p = 12;
  for (int a = 0; a < 12; ++a)
    for (int c = a; c < 12; ++c) {
      const float coef = (a == c) ? wo[a] * wo[a] : 2.f * wo[a] * wo[c];
      quad += coef * S[p++];
    }
  const float mean = (s1w + be * Mf) / Mf;
  const float ey2 = (quad + 2.f * be * s1w + be * be * Mf) / Mf;
  const float var = ey2 - mean * mean;
  const float r = gamma[o] * rsqrtf(var + BN_EPS);
  sc[o] = r;
  sh[o] = beta[o] - mean * r;
}

// per point: build 20x12 edge features in LDS, 64 channels, bn+relu, max over K
__global__ __launch_bounds__(64) void mt_graph_apply(
    const float* __restrict__ pts, const int* __restrict__ idx,
    const float* __restrict__ w, const float* __restrict__ bias,
    const float* __restrict__ sc, const float* __restrict__ sh,
    float* __restrict__ out) {
  __shared__ float feat[KNN][12];
  const int row = blockIdx.x;          // b*N + n
  const int b = row / N_PTS;
  const int n = row % N_PTS;
  const int t = threadIdx.x;
  const float* c3 = pts + (size_t)row * 3;
  if (t < KNN) {
    const int j = idx[(size_t)row * KNN + t];
    const float* n3 = pts + ((size_t)b * N_PTS + j) * 3;
    float f[12];
    make_feat(c3, n3, f);
#pragma unroll
    for (int q = 0; q < 12; ++q) feat[t][q] = f[q];
  }
  __syncthreads();
  const int o = t;
  float wl[12];
#pragma unroll
  for (int q = 0; q < 12; ++q) wl[q] = w[o * 12 + q];
  const float be = bias[o], s = sc[o], h = sh[o];
  float m = NEG_INF;
  for (int k = 0; k < KNN; ++k) {
    float y = be;
#pragma unroll
    for (int q = 0; q < 12; ++q) y += wl[q] * feat[k][q];
    y = fmaxf(y * s + h, 0.f);
    m = fmaxf(m, y);
  }
  out[((size_t)b * 64 + o) * N_PTS + n] = m;
}

// ---------------------------------------------------------------------------
extern "C" void kernel_launch(void* const* d_in, const int* in_sizes, int n_in,
                              void* d_out, int out_size, void* d_ws,
                              size_t ws_size, hipStream_t stream) {
  (void)in_sizes; (void)n_in; (void)out_size; (void)ws_size;
  const float* x    = (const float*)d_in[0];
  const float* tw1  = (const float*)d_in[1];
  const float* tb1  = (const float*)d_in[2];
  const float* tg1  = (const float*)d_in[3];
  const float* tbe1 = (const float*)d_in[4];
  const float* tw2  = (const float*)d_in[5];
  const float* tb2  = (const float*)d_in[6];
  const float* tg2  = (const float*)d_in[7];
  const float* tbe2 = (const float*)d_in[8];
  const float* tw3  = (const float*)d_in[9];
  const float* tb3  = (const float*)d_in[10];
  const float* tg3  = (const float*)d_in[11];
  const float* tbe3 = (const float*)d_in[12];
  const float* fw1  = (const float*)d_in[13];
  const float* fb1  = (const float*)d_in[14];
  const float* tg4  = (const float*)d_in[15];
  const float* tbe4 = (const float*)d_in[16];
  const float* fw2  = (const float*)d_in[17];
  const float* fb2  = (const float*)d_in[18];
  const float* tg5  = (const float*)d_in[19];
  const float* tbe5 = (const float*)d_in[20];
  const float* fw3  = (const float*)d_in[21];
  const float* fb3  = (const float*)d_in[22];
  const float* cw1  = (const float*)d_in[23];
  const float* cb1  = (const float*)d_in[24];
  const float* g1p  = (const float*)d_in[25];
  const float* be1  = (const float*)d_in[26];
  const float* cw2  = (const float*)d_in[27];
  const float* cb2  = (const float*)d_in[28];
  const float* g2p  = (const float*)d_in[29];
  const float* be2  = (const float*)d_in[30];
  const float* cw3  = (const float*)d_in[31];
  const float* cb3  = (const float*)d_in[32];
  const float* g3p  = (const float*)d_in[33];
  const float* be3  = (const float*)d_in[34];
  const float* cw4  = (const float*)d_in[35];
  const float* cb4  = (const float*)d_in[36];
  const float* g4p  = (const float*)d_in[37];
  const float* be4  = (const float*)d_in[38];
  const float* cw5  = (const float*)d_in[39];
  const float* cb5  = (const float*)d_in[40];
  const float* g5p  = (const float*)d_in[41];
  const float* be5  = (const float*)d_in[42];
  const float* cw6  = (const float*)d_in[43];
  const float* cb6  = (const float*)d_in[44];

  float* ws  = (float*)d_ws;
  float* A   = ws + OFF_A;
  float* Bb  = ws + OFF_B;
  float* Cc  = ws + OFF_C;
  float* pts = ws + OFF_PTS;
  float* xx  = ws + OFF_XX;
  float* pool = ws + OFF_POOL;
  float* f1z = ws + OFF_F1Z;
  float* f1  = ws + OFF_F1;
  float* f2z = ws + OFF_F2Z;
  float* f2  = ws + OFF_F2;
  float* mat = ws + OFF_MAT;
  float* stat = ws + OFF_STAT;
  float* sc  = ws + OFF_SC;
  float* sh  = ws + OFF_SH;
  float* S   = ws + OFF_S1;
  int*   idx = (int*)(ws + OFF_IDX);
  float* outF = (float*)d_out;

  const int BN = B_SZ * N_PTS;  // 32768

  // ---- TNet ----
  mt_conv_small<<<8192, 256, 0, stream>>>(x, tw1, tb1, A, 3, 64);
  mt_stats_finalize<<<64, 256, 0, stream>>>(A, tg1, tbe1, sc, sh, 64);
  mt_bn_relu<<<8192, 256, 0, stream>>>(A, sc, sh, 64);

  mt_gemm_store<<<dim3(64, 8, 8), 128, 0, stream>>>(tw2, A, tb2, Bb, 64, 128);
  mt_stats_finalize<<<128, 256, 0, stream>>>(Bb, tg2, tbe2, sc, sh, 128);
  mt_bn_relu<<<16384, 256, 0, stream>>>(Bb, sc, sh, 128);

  hipMemsetAsync(stat, 0, 2 * 1024 * sizeof(float), stream);
  mt_gemm_stats<<<dim3(64, 8), 128, 0, stream>>>(tw3, Bb, tb3, stat, 128, 1024);
  mt_finalize_sums<<<4, 256, 0, stream>>>(stat, tg3, tbe3, sc, sh, 1024,
                                          (float)BN);
  mt_gemm_reduce_max<<<dim3(64, 8), 128, 0, stream>>>(tw3, Bb, tb3, sc, sh, 1,
                                                      1, pool, 128, 1024);

  mt_fc<<<16, 256, 0, stream>>>(pool, fw1, fb1, f1z, 1024, 512, 0);
  mt_bn1d_relu<<<2, 256, 0, stream>>>(f1z, tg4, tbe4, f1, 512);
  mt_fc<<<8, 256, 0, stream>>>(f1, fw2, fb2, f2z, 512, 256, 0);
  mt_bn1d_relu<<<1, 256, 0, stream>>>(f2z, tg5, tbe5, f2, 256);
  mt_fc<<<1, 128, 0, stream>>>(f2, fw3, fb3, mat, 256, 9, 1);

  // ---- transform + KNN + edge features ----
  mt_transform<<<128, 256, 0, stream>>>(x, mat, pts, xx);
  mt_knn<<<BN, 128, 0, stream>>>(pts, xx, idx);

  hipMemsetAsync(S, 0, 90 * sizeof(float), stream);
  mt_feat_stats<<<256, 256, 0, stream>>>(pts, idx, S);
  mt_graph_finalize<<<1, 64, 0, stream>>>(cw1, cb1, g1p, be1, S, sc, sh);
  mt_graph_apply<<<BN, 64, 0, stream>>>(pts, idx, cw1, cb1, sc, sh, A);

  // ---- conv stack ----
  mt_gemm_store<<<dim3(64, 4, 8), 128, 0, stream>>>(cw2, A, cb2, Cc, 64, 64);
  mt_stats_finalize<<<64, 256, 0, stream>>>(Cc, g2p, be2, sc, sh, 64);
  mt_bn_relu<<<8192, 256, 0, stream>>>(Cc, sc, sh, 64);

  mt_gemm_store<<<dim3(64, 4, 8), 128, 0, stream>>>(cw3, Cc, cb3, A, 64, 64);
  mt_stats_finalize<<<64, 256, 0, stream>>>(A, g3p, be3, sc, sh, 64);
  mt_bn_relu<<<8192, 256, 0, stream>>>(A, sc, sh, 64);

  mt_gemm_store<<<dim3(64, 4, 8), 128, 0, stream>>>(cw4, A, cb4, Cc, 64, 64);
  mt_stats_finalize<<<64, 256, 0, stream>>>(Cc, g4p, be4, sc, sh, 64);
  mt_bn_relu<<<8192, 256, 0, stream>>>(Cc, sc, sh, 64);

  mt_gemm_store<<<dim3(64, 8, 8), 128, 0, stream>>>(cw5, Cc, cb5, Bb, 64, 128);
  mt_stats_finalize<<<128, 256, 0, stream>>>(Bb, g5p, be5, sc, sh, 128);
  mt_bn_relu<<<16384, 256, 0, stream>>>(Bb, sc, sh, 128);

  // conv6 (no BN/relu) fused with global max over N -> (8,1024) output
  mt_gemm_reduce_max<<<dim3(64, 8), 128, 0, stream>>>(
      cw6, Bb, cb6, nullptr, nullptr, 0, 0, outF, 128, 1024);
}